// GNNCore_85633057948392
// MI455X (gfx1250) — compile-verified
//
#include <hip/hip_runtime.h>
#include <hip/hip_bf16.h>

typedef __attribute__((ext_vector_type(2))) float v2f;
typedef __attribute__((ext_vector_type(8))) float v8f;

#define NEG_SLOPE 0.01f

// ---------------- degree / normalization ----------------

__global__ __launch_bounds__(256) void k_init_deg(float* __restrict__ deg, int n) {
  int i = blockIdx.x * blockDim.x + threadIdx.x;
  if (i < n) deg[i] = 1.0f;  // self-loop
}

__global__ __launch_bounds__(256) void k_count_deg(const int* __restrict__ dst,
                                                   float* __restrict__ deg, int e) {
  int i = blockIdx.x * blockDim.x + threadIdx.x;
  if (i < e) atomicAdd(&deg[dst[i]], 1.0f);
}

__global__ __launch_bounds__(256) void k_rsqrt(float* __restrict__ deg, int n) {
  int i = blockIdx.x * blockDim.x + threadIdx.x;
  if (i < n) deg[i] = rsqrtf(deg[i]);
}

// ---------------- fp32 WMMA GEMM: H[M,Nout] = act(X)[M,64] @ W[64,Nout] -------
// One wave32 per 16x16 output tile; K=64 via 16x V_WMMA_F32_16X16X4_F32.
// A layout (16x4 f32): lanes 0-15: M=lane, VGPR0/1 = K=0/1; lanes 16-31: K=2/3.
// B/C/D: row striped across lanes within a VGPR (C: VGPRv -> M=v (+8 for hi half)).

__global__ __launch_bounds__(32) void k_gemm_wmma(const float* __restrict__ X,
                                                  const float* __restrict__ W,
                                                  float* __restrict__ H,
                                                  int M, int Nout, int leaky) {
  const int K = 64;
  const int lane = threadIdx.x;        // 0..31, wave32
  const int m    = lane & 15;          // row-in-tile / col-in-tile
  const int khi  = (lane >> 4) << 1;   // 0 or 2 (K sub-offset per half-wave)

  const int row0 = blockIdx.x << 4;
  const int col0 = blockIdx.y << 4;

  int arow = row0 + m;
  if (arow >= M) arow = M - 1;         // clamp loads: EXEC must stay all-ones for WMMA
  const float* __restrict__ xr = X + (long long)arow * K;
  const float* __restrict__ wc = W + col0 + m;

  v8f c = {};
#pragma unroll
  for (int k0 = 0; k0 < K; k0 += 4) {
    float a0 = xr[k0 + khi + 0];
    float a1 = xr[k0 + khi + 1];
    if (leaky) {  // fused LeakyReLU on the A operand (layer-2 input)
      a0 = (a0 > 0.0f) ? a0 : NEG_SLOPE * a0;
      a1 = (a1 > 0.0f) ? a1 : NEG_SLOPE * a1;
    }
    v2f a, b;
    a.x = a0;
    a.y = a1;
    b.x = wc[(k0 + khi + 0) * Nout];
    b.y = wc[(k0 + khi + 1) * Nout];
    // 8 args: (neg_a, A, neg_b, B, c_mod, C, reuse_a, reuse_b)
    c = __builtin_amdgcn_wmma_f32_16x16x4_f32(false, a, false, b, (short)0, c,
                                              false, false);
  }

  const int hi8 = (lane >> 4) << 3;    // 0 or 8
#pragma unroll
  for (int v = 0; v < 8; ++v) {
    int r = row0 + hi8 + v;
    if (r < M) H[(long long)r * Nout + col0 + m] = c[v];
  }
}

// ---------------- aggregation ----------------
// A[i,:] = bias + dinv[i]^2 * H[i,:]   (self-loop term + bias)
__global__ __launch_bounds__(256) void k_agg_init(const float* __restrict__ H,
                                                  const float* __restrict__ dinv,
                                                  const float* __restrict__ bias,
                                                  float* __restrict__ A,
                                                  int n, int C, int tshift) {
  long long gid = (long long)blockIdx.x * blockDim.x + threadIdx.x;
  if (gid >= ((long long)n << tshift)) return;
  int i = (int)(gid >> tshift);
  int t = (int)(gid & ((1 << tshift) - 1));
  float di = dinv[i];
  float w  = di * di;
  float4 h = *((const float4*)(H + (long long)i * C) + t);
  float4 b = *((const float4*)bias + t);
  float4 r = make_float4(b.x + w * h.x, b.y + w * h.y,
                         b.z + w * h.z, b.w + w * h.w);
  *((float4*)(A + (long long)i * C) + t) = r;
}

// per-edge: A[dst,:] += dinv[src]*dinv[dst] * H[src,:]   (C/4 lanes per edge)
__global__ __launch_bounds__(256) void k_agg_edges(const float* __restrict__ H,
                                                   const float* __restrict__ dinv,
                                                   const int* __restrict__ src,
                                                   const int* __restrict__ dst,
                                                   float* __restrict__ A,
                                                   int e, int C, int tshift) {
  long long gid = (long long)blockIdx.x * blockDim.x + threadIdx.x;
  if (gid >= ((long long)e << tshift)) return;
  int ed = (int)(gid >> tshift);
  int t  = (int)(gid & ((1 << tshift) - 1));
  int s = src[ed];
  int d = dst[ed];
  float w = dinv[s] * dinv[d];
  float4 h = *((const float4*)(H + (long long)s * C) + t);
  float* o = A + (long long)d * C + (t << 2);
  atomicAdd(o + 0, w * h.x);
  atomicAdd(o + 1, w * h.y);
  atomicAdd(o + 2, w * h.z);
  atomicAdd(o + 3, w * h.w);
}

// ---------------- launch ----------------

extern "C" void kernel_launch(void* const* d_in, const int* in_sizes, int n_in,
                              void* d_out, int out_size, void* d_ws, size_t ws_size,
                              hipStream_t stream) {
  (void)n_in; (void)out_size; (void)ws_size;

  const float* x  = (const float*)d_in[0];
  const int*   ei = (const int*)  d_in[1];
  const float* W1 = (const float*)d_in[2];
  const float* b1 = (const float*)d_in[3];
  const float* W2 = (const float*)d_in[4];
  const float* b2 = (const float*)d_in[5];
  float* out = (float*)d_out;

  const int DIN  = 64;
  const int N    = in_sizes[0] / DIN;
  const int E    = in_sizes[1] / 2;
  const int DOUT = in_sizes[5];  // 32

  const int* src = ei;       // edge_index[0]
  const int* dst = ei + E;   // edge_index[1]

  // workspace layout (floats): dinv | H1 (reused as H2) | A1
  float* dinv = (float*)d_ws;
  int Nr = ((N + 255) / 256) * 256;
  float* H1 = dinv + Nr;
  float* A1 = H1 + (size_t)N * DIN;
  float* H2 = H1;  // H1 dead after layer-1 aggregation

  const int TB = 256;

  // normalization: deg = 1 + sum over incoming edges; dinv = rsqrt(deg)
  k_init_deg <<<(N + TB - 1) / TB, TB, 0, stream>>>(dinv, N);
  k_count_deg<<<(E + TB - 1) / TB, TB, 0, stream>>>(dst, dinv, E);
  k_rsqrt    <<<(N + TB - 1) / TB, TB, 0, stream>>>(dinv, N);

  const int Mtiles = (N + 15) / 16;

  // layer 1: H1 = x @ W1 ; A1 = segsum(norm * H1[src]) + b1
  k_gemm_wmma<<<dim3(Mtiles, DIN / 16), 32, 0, stream>>>(x, W1, H1, N, DIN, 0);
  {
    long long tot = (long long)N << 4;  // N * 64/4
    k_agg_init<<<(int)((tot + TB - 1) / TB), TB, 0, stream>>>(H1, dinv, b1, A1, N, DIN, 4);
    tot = (long long)E << 4;
    k_agg_edges<<<(int)((tot + TB - 1) / TB), TB, 0, stream>>>(H1, dinv, src, dst, A1, E, DIN, 4);
  }

  // layer 2: H2 = leakyrelu(A1) @ W2 (fused) ; out = segsum(norm * H2[src]) + b2
  k_gemm_wmma<<<dim3(Mtiles, DOUT / 16), 32, 0, stream>>>(A1, W2, H2, N, DOUT, 1);
  {
    long long tot = (long long)N << 3;  // N * 32/4
    k_agg_init<<<(int)((tot + TB - 1) / TB), TB, 0, stream>>>(H2, dinv, b2, out, N, DOUT, 3);
    tot = (long long)E << 3;
    k_agg_edges<<<(int)((tot + TB - 1) / TB), TB, 0, stream>>>(H2, dinv, src, dst, out, E, DOUT, 3);
  }
}